// SplatsVAE_45518063403305
// MI455X (gfx1250) — compile-verified
//
#include <hip/hip_runtime.h>
#include <hip/hip_bf16.h>

// CDNA5 / gfx1250, wave32. Separable-Gaussian reformulation turns the splat
// render into per-batch GEMMs (256x256x2048) done with V_WMMA_F32_16X16X4_F32.
// Factor matrices (16 MB) live in the 192 MB L2, so the WMMA loop streams from L2.
// Dual accumulators break the D->C WMMA dependency chain for XDL-pipe ILP.

typedef __attribute__((ext_vector_type(2))) float v2f;
typedef __attribute__((ext_vector_type(8))) float v8f;

#define BATCH 4
#define DIN   64
#define HID   512
#define KSPL  2048
#define NGEOM 8192   // K*4
#define NAPP  4096   // K*2
#define IMG   256

__device__ __forceinline__ float sigmoidf_(float x) { return 1.0f / (1.0f + __expf(-x)); }

__device__ __forceinline__ v8f wmma_f32(v2f a, v2f b, v8f c) {
    return __builtin_amdgcn_wmma_f32_16x16x4_f32(false, a, false, b,
                                                 (short)0, c, false, false);
}

// ---------------- MLP layer 1 (both MLPs): h1 = relu(z @ w1 + b1) ----------
// 1024 threads: thread t -> (which MLP, hidden unit h). 4 batch rows per thread.
__global__ void k_layer1(const float* __restrict__ z,
                         const float* __restrict__ gw1, const float* __restrict__ gb1,
                         const float* __restrict__ aw1, const float* __restrict__ ab1,
                         float* __restrict__ h1g, float* __restrict__ h1a)
{
    int t = blockIdx.x * blockDim.x + threadIdx.x;   // 0..1023
    int which = t >> 9;
    int h = t & (HID - 1);
    const float* w    = which ? aw1 : gw1;
    const float* bias = which ? ab1 : gb1;
    float* out        = which ? h1a : h1g;
    float a0 = 0.f, a1 = 0.f, a2 = 0.f, a3 = 0.f;
    for (int d = 0; d < DIN; ++d) {
        float wv = w[d * HID + h];
        a0 += z[0 * DIN + d] * wv;
        a1 += z[1 * DIN + d] * wv;
        a2 += z[2 * DIN + d] * wv;
        a3 += z[3 * DIN + d] * wv;
    }
    float bb = bias[h];
    out[0 * HID + h] = fmaxf(a0 + bb, 0.f);
    out[1 * HID + h] = fmaxf(a1 + bb, 0.f);
    out[2 * HID + h] = fmaxf(a2 + bb, 0.f);
    out[3 * HID + h] = fmaxf(a3 + bb, 0.f);
}

// ---------------- MLP layer 2: h2 = relu(h1 @ w2 + b2), padded to 16 rows ---
__global__ void k_layer2(const float* __restrict__ h1g, const float* __restrict__ h1a,
                         const float* __restrict__ gw2, const float* __restrict__ gb2,
                         const float* __restrict__ aw2, const float* __restrict__ ab2,
                         float* __restrict__ h2g, float* __restrict__ h2a)
{
    int t = blockIdx.x * blockDim.x + threadIdx.x;   // 0..1023
    int which = t >> 9;
    int h = t & (HID - 1);
    const float* hin  = which ? h1a : h1g;
    const float* w    = which ? aw2 : gw2;
    const float* bias = which ? ab2 : gb2;
    float* out        = which ? h2a : h2g;
    float a0 = 0.f, a1 = 0.f, a2 = 0.f, a3 = 0.f;
    for (int d = 0; d < HID; ++d) {
        float wv = w[d * HID + h];
        a0 += hin[0 * HID + d] * wv;
        a1 += hin[1 * HID + d] * wv;
        a2 += hin[2 * HID + d] * wv;
        a3 += hin[3 * HID + d] * wv;
    }
    float bb = bias[h];
    out[0 * HID + h] = fmaxf(a0 + bb, 0.f);
    out[1 * HID + h] = fmaxf(a1 + bb, 0.f);
    out[2 * HID + h] = fmaxf(a2 + bb, 0.f);
    out[3 * HID + h] = fmaxf(a3 + bb, 0.f);
    // zero the M-padding rows (4..15) so WMMA sees a clean 16-row A matrix
    for (int m = BATCH; m < 16; ++m) out[m * HID + h] = 0.f;
}

// ---------------- MLP layer 3 via f32 WMMA: out = h2(16x512) @ w3 + b3 ------
// 768 tiles of N=16 (512 geometry + 256 appearance). One wave per tile,
// K=512 in steps of 8 with two independent accumulators (XDL ILP).
__global__ void k_layer3(const float* __restrict__ h2g, const float* __restrict__ h2a,
                         const float* __restrict__ gw3, const float* __restrict__ gb3,
                         const float* __restrict__ aw3, const float* __restrict__ ab3,
                         float* __restrict__ geomO, float* __restrict__ appO)
{
    int lane = threadIdx.x & 31;
    int wave = threadIdx.x >> 5;
    int g = blockIdx.x * 8 + wave;                    // 0..767 (wave-uniform)
    const float* A; const float* W; const float* bias; float* out; int N; int n0;
    if (g < NGEOM / 16) { A = h2g; W = gw3; bias = gb3; out = geomO; N = NGEOM; n0 = g * 16; }
    else                { A = h2a; W = aw3; bias = ab3; out = appO;  N = NAPP;  n0 = (g - NGEOM / 16) * 16; }

    int m  = lane & 15;    // A row (M) / B column (N) owned by this lane
    int hh = lane >> 4;    // half-wave selects K pair {0,1} vs {2,3}

    v8f c0 = {}, c1 = {};
    for (int k = 0; k < HID; k += 8) {
        int k0 = k + 2 * hh;
        int k1 = k + 4 + 2 * hh;
        v2f a0 = *(const v2f*)(A + m * HID + k0);
        v2f a1 = *(const v2f*)(A + m * HID + k1);
        v2f b0, b1;
        b0.x = W[(k0)     * N + n0 + m];
        b0.y = W[(k0 + 1) * N + n0 + m];
        b1.x = W[(k1)     * N + n0 + m];
        b1.y = W[(k1 + 1) * N + n0 + m];
        c0 = wmma_f32(a0, b0, c0);
        c1 = wmma_f32(a1, b1, c1);
    }
    v8f c = c0 + c1;
    int n = n0 + m;
    float bv = bias[n];
#pragma unroll
    for (int r = 0; r < 8; ++r) {
        int row = r + 8 * hh;                  // D layout: M = r + 8*(lane/16)
        if (row < BATCH) out[row * N + n] = c[r] + bv;
    }
}

// ---------------- splat assembly -------------------------------------------
__global__ void k_splats(const float* __restrict__ geomO, const float* __restrict__ appO,
                         float* __restrict__ splats)
{
    int t = blockIdx.x * blockDim.x + threadIdx.x;   // 0..8191
    int b = t >> 11;
    int k = t & (KSPL - 1);
    float g0 = geomO[b * NGEOM + k * 4 + 0];
    float g1 = geomO[b * NGEOM + k * 4 + 1];
    float g2 = geomO[b * NGEOM + k * 4 + 2];
    float g3 = geomO[b * NGEOM + k * 4 + 3];
    float a0 = appO[b * NAPP + k * 2 + 0];
    float a1 = appO[b * NAPP + k * 2 + 1];
    float r  = (g3 > 20.f) ? g3 : log1pf(__expf(g3));   // softplus
    float* s = splats + ((size_t)b * KSPL + k) * 8;
    s[0] = g0; s[1] = g1; s[2] = g2; s[3] = r;
    s[4] = sigmoidf_(a0);   // intensity
    s[5] = a1;              // raw opacity logit (reference stores it raw)
    s[6] = 0.f; s[7] = 0.f;
}

// ---------------- separable Gaussian factors -------------------------------
// colF[b][i][k] = sigmoid(opac)*inten * exp(-(grid[i]-yc)^2 / 2r^2)
// rowF[b][j][k] =                        exp(-(grid[j]-xc)^2 / 2r^2)
// Layout [b][pos][k] so each WMMA lane later loads a contiguous float2 in k.
// 256 blocks (2048 waves): block -> (b, k-chunk of 256, 32-position group).
__global__ void k_factors(const float* __restrict__ splats,
                          float* __restrict__ colF, float* __restrict__ rowF)
{
    int b      = blockIdx.x >> 6;           // 0..3
    int sub    = blockIdx.x & 63;
    int kchunk = sub & 7;                   // 8 chunks of 256 splats
    int pbase  = (sub >> 3) * 32;           // 8 groups of 32 positions
    int k      = kchunk * 256 + threadIdx.x;

    const float* s = splats + ((size_t)b * KSPL + k) * 8;
    float xc = s[0], yc = s[1];
    float r  = fmaxf(s[3], 0.01f);
    float w  = sigmoidf_(s[5]) * s[4];
    float inv = 1.0f / (2.0f * r * r);
    float* cb = colF + (size_t)b * IMG * KSPL + k;
    float* rb = rowF + (size_t)b * IMG * KSPL + k;
#pragma unroll 4
    for (int p = pbase; p < pbase + 32; ++p) {
        float gc = -1.0f + (2.0f / 255.0f) * p;      // jnp.linspace(-1,1,256)
        float dy = gc - yc;
        float dx = gc - xc;
        cb[(size_t)p * KSPL] = w * __expf(-dy * dy * inv);
        rb[(size_t)p * KSPL] =     __expf(-dx * dx * inv);
    }
}

// ---------------- render GEMM: img[b] = colF[b] @ rowF[b]^T (K=2048) -------
// 1024 waves, one 16x16 output tile each, K step 8 with dual accumulators;
// global_prefetch keeps the WGP cache one line-group ahead; clip epilogue.
__global__ void k_render(const float* __restrict__ colF, const float* __restrict__ rowF,
                         float* __restrict__ img)
{
    int lane = threadIdx.x & 31;
    int wave = threadIdx.x >> 5;
    int gwv = blockIdx.x * 8 + wave;                 // 0..1023 (wave-uniform)
    int b   = gwv >> 8;
    int t   = gwv & 255;
    int i0  = (t >> 4) * 16;
    int j0  = (t & 15) * 16;
    int m   = lane & 15;
    int hh  = lane >> 4;

    const float* Ap = colF + (size_t)b * IMG * KSPL + (size_t)(i0 + m) * KSPL;
    const float* Bp = rowF + (size_t)b * IMG * KSPL + (size_t)(j0 + m) * KSPL;

    v8f c0 = {}, c1 = {};
    for (int k = 0; k < KSPL; k += 8) {
        if ((k & 127) == 0 && k + 128 < KSPL) {      // uniform branch; EXEC stays full
            __builtin_prefetch(Ap + k + 128, 0, 1);  // -> global_prefetch_b8
            __builtin_prefetch(Bp + k + 128, 0, 1);
        }
        int k0 = k + 2 * hh;
        int k1 = k + 4 + 2 * hh;
        v2f a0 = *(const v2f*)(Ap + k0);   // A[M=i][k'] = colF[b][i][k']
        v2f a1 = *(const v2f*)(Ap + k1);
        v2f b0 = *(const v2f*)(Bp + k0);   // B[k'][N=j] = rowF[b][j][k']
        v2f b1 = *(const v2f*)(Bp + k1);
        c0 = wmma_f32(a0, b0, c0);
        c1 = wmma_f32(a1, b1, c1);
    }
    v8f c = c0 + c1;
    float* outb = img + (size_t)b * IMG * IMG;
#pragma unroll
    for (int r = 0; r < 8; ++r) {
        int row = i0 + r + 8 * hh;
        int col = j0 + m;
        float v = fminf(fmaxf(c[r], 0.0f), 1.0f);
        outb[row * IMG + col] = v;
    }
}

// ---------------- host-side launch ------------------------------------------
extern "C" void kernel_launch(void* const* d_in, const int* in_sizes, int n_in,
                              void* d_out, int out_size, void* d_ws, size_t ws_size,
                              hipStream_t stream) {
    const float* z   = (const float*)d_in[0];
    const float* gw1 = (const float*)d_in[1];
    const float* gb1 = (const float*)d_in[2];
    const float* gw2 = (const float*)d_in[3];
    const float* gb2 = (const float*)d_in[4];
    const float* gw3 = (const float*)d_in[5];
    const float* gb3 = (const float*)d_in[6];
    const float* aw1 = (const float*)d_in[7];
    const float* ab1 = (const float*)d_in[8];
    const float* aw2 = (const float*)d_in[9];
    const float* ab2 = (const float*)d_in[10];
    const float* aw3 = (const float*)d_in[11];
    const float* ab3 = (const float*)d_in[12];

    float* out      = (float*)d_out;
    float* rendered = out;                                   // 4*1*256*256
    float* splats   = out + (size_t)BATCH * IMG * IMG;       // 4*2048*8

    // workspace carve-up (floats): ~17 MB total
    float* ws    = (float*)d_ws;
    float* h1g   = ws;                    // 4*512
    float* h1a   = h1g  + BATCH * HID;
    float* h2g   = h1a  + BATCH * HID;    // 16*512 (padded)
    float* h2a   = h2g  + 16 * HID;
    float* geomO = h2a  + 16 * HID;       // 4*8192
    float* appO  = geomO + BATCH * NGEOM; // 4*4096
    float* colF  = appO  + BATCH * NAPP;  // 4*256*2048
    float* rowF  = colF  + (size_t)BATCH * IMG * KSPL;

    k_layer1 <<<4,   256, 0, stream>>>(z, gw1, gb1, aw1, ab1, h1g, h1a);
    k_layer2 <<<4,   256, 0, stream>>>(h1g, h1a, gw2, gb2, aw2, ab2, h2g, h2a);
    k_layer3 <<<96,  256, 0, stream>>>(h2g, h2a, gw3, gb3, aw3, ab3, geomO, appO);
    k_splats <<<32,  256, 0, stream>>>(geomO, appO, splats);
    k_factors<<<256, 256, 0, stream>>>(splats, colF, rowF);
    k_render <<<128, 256, 0, stream>>>(colF, rowF, rendered);
}